// GATEConv_25821343383694
// MI455X (gfx1250) — compile-verified
//
#include <hip/hip_runtime.h>

#define NN 4096
#define DIM 128
#define EDIM 8
#define NEG_SLOPE 0.2f
#define NEG_INF -1.0e9f

typedef __attribute__((ext_vector_type(2))) float v2f;
typedef __attribute__((ext_vector_type(8))) float v8f;

// ---------------------------------------------------------------------------
// Kernel 1: per-row projections + attention score scalars
//   h_dst[i][:]  = node[i] @ W_dst + b_dst        (kept, needed for out GEMM)
//   s_src[i]     = (node[i] @ W_src + b_src) . a_s
//   s_dst[i]     = h_dst[i] . a_d
//   w_ea[0..7]   = W_edge @ a_e ;  w_ea[8] = b_edge . a_e   (block 0 only)
// ---------------------------------------------------------------------------
__global__ __launch_bounds__(128)
void gat_prep(const float* __restrict__ node,
              const float* __restrict__ Wsrc, const float* __restrict__ bsrc,
              const float* __restrict__ Wdst, const float* __restrict__ bdst,
              const float* __restrict__ Wedge, const float* __restrict__ bedge,
              const float* __restrict__ a,
              float* __restrict__ h_dst,
              float* __restrict__ s_src, float* __restrict__ s_dst,
              float* __restrict__ w_ea)
{
    __shared__ float x[DIM];
    __shared__ float red[DIM];
    const int i = blockIdx.x;
    const int o = threadIdx.x;

    if (i == 0 && o <= EDIM) {
        float acc = 0.f;
        if (o < EDIM) {
            for (int c = 0; c < DIM; ++c) acc += Wedge[o * DIM + c] * a[2 * DIM + c];
        } else {
            for (int c = 0; c < DIM; ++c) acc += bedge[c] * a[2 * DIM + c];
        }
        w_ea[o] = acc;
    }

    x[o] = node[(size_t)i * DIM + o];
    __syncthreads();

    float hs = bsrc[o], hd = bdst[o];
#pragma unroll 4
    for (int k = 0; k < DIM; ++k) {
        float xk = x[k];
        hs = fmaf(xk, Wsrc[k * DIM + o], hs);
        hd = fmaf(xk, Wdst[k * DIM + o], hd);
    }
    h_dst[(size_t)i * DIM + o] = hd;

    red[o] = hs * a[o];
    __syncthreads();
    for (int s = 64; s > 0; s >>= 1) {
        if (o < s) red[o] += red[o + s];
        __syncthreads();
    }
    if (o == 0) s_src[i] = red[0];
    __syncthreads();
    red[o] = hd * a[DIM + o];
    __syncthreads();
    for (int s = 64; s > 0; s >>= 1) {
        if (o < s) red[o] += red[o + s];
        __syncthreads();
    }
    if (o == 0) s_dst[i] = red[0];
}

// ---------------------------------------------------------------------------
// Kernel 2: fused edge-score + masked LeakyReLU softmax + attn @ h_dst (WMMA)
// One block = 16 attention rows. 256 threads = 8 waves.
// ---------------------------------------------------------------------------
#define KC 64
#define HB_S (DIM + 4)  // 132 floats; 528B row -> 16B aligned, bank-spread
#define PA_S (KC + 4)   // 68

__global__ __launch_bounds__(256)
void gat_attn(const float* __restrict__ adj,
              const float* __restrict__ edge,
              const float* __restrict__ h_dst,
              const float* __restrict__ s_src,
              const float* __restrict__ s_dst,
              const float* __restrict__ w_ea,
              float* __restrict__ out,
              float* __restrict__ attn)
{
    __shared__ float hB[KC * HB_S];
    __shared__ float pA[16 * PA_S];
    __shared__ float redM[16][17];
    __shared__ float redL[16][17];
    __shared__ float rowM[16];
    __shared__ float rowInvL[16];
    __shared__ float we[EDIM + 1];

    const int tid = threadIdx.x;
    const int i0 = blockIdx.x * 16;

    if (tid <= EDIM) we[tid] = w_ea[tid];
    __syncthreads();

    const int li = tid >> 4;  // local row 0..15
    const int jl = tid & 15;  // column phase within row

    const size_t rowBase = (size_t)(i0 + li) * NN;
    const float ss = s_src[i0 + li];
    const float w0 = we[0], w1 = we[1], w2 = we[2], w3 = we[3];
    const float w4 = we[4], w5 = we[5], w6 = we[6], w7 = we[7];
    const float eb = we[8];

    // ---- pass 1: stream edge_features, build masked logits + online (m,l) ----
    float m = -3.0e38f, l = 0.f;
#pragma unroll 2
    for (int j = jl; j < NN; j += 16) {
        const float4* ep = (const float4*)(edge + (rowBase + j) * EDIM);
        float4 e0 = ep[0];
        float4 e1 = ep[1];
        float es = eb;
        es = fmaf(e0.x, w0, es); es = fmaf(e0.y, w1, es);
        es = fmaf(e0.z, w2, es); es = fmaf(e0.w, w3, es);
        es = fmaf(e1.x, w4, es); es = fmaf(e1.y, w5, es);
        es = fmaf(e1.z, w6, es); es = fmaf(e1.w, w7, es);
        float logit = ss + s_dst[j] + es;
        logit = (logit >= 0.f) ? logit : NEG_SLOPE * logit;
        logit = (adj[rowBase + j] > 0.f) ? logit : NEG_INF;
        attn[rowBase + j] = logit;  // raw logit scratch, overwritten in pass 2
        float mn = fmaxf(m, logit);
        l = l * __expf(m - mn) + __expf(logit - mn);
        m = mn;
    }
    redM[li][jl] = m;
    redL[li][jl] = l;
    __syncthreads();
    if (tid < 16) {
        float M = -3.0e38f;
        for (int t = 0; t < 16; ++t) M = fmaxf(M, redM[tid][t]);
        float L = 0.f;
        for (int t = 0; t < 16; ++t) L += redL[tid][t] * __expf(redM[tid][t] - M);
        rowM[tid] = M;
        rowInvL[tid] = 1.0f / L;
    }
    __threadfence_block();
    __syncthreads();

    // ---- pass 2: normalize -> attn output, and out = attn @ h_dst via WMMA ----
    const int wv = tid >> 5;      // wave -> 16-column slab of out
    const int lane = tid & 31;
    const int half = lane >> 4;   // 0/1 lane half per WMMA layout
    const int ln = lane & 15;

    v8f acc = {};
    for (int j0 = 0; j0 < NN; j0 += KC) {
        // stage h_dst chunk (KC x DIM) into LDS
        const float4* src = (const float4*)(h_dst + (size_t)j0 * DIM);
        for (int q = tid; q < KC * (DIM / 4); q += 256) {
            int r = q >> 5;   // 32 float4 per row
            int c4 = q & 31;
            float4 v = src[r * 32 + c4];
            *(float4*)&hB[r * HB_S + c4 * 4] = v;
        }
        // normalize this chunk of logits (each thread re-reads its own writes)
        {
            float M = rowM[li], invL = rowInvL[li];
#pragma unroll
            for (int s = 0; s < KC / 16; ++s) {
                int jj = jl + 16 * s;
                size_t g = rowBase + j0 + jj;
                float x = attn[g];
                float p = (x > -1.0e8f) ? __expf(x - M) * invL : 0.f;
                attn[g] = p;
                pA[li * PA_S + jj] = p;
            }
        }
        __syncthreads();

        // 16x16x4 fp32 WMMA over the K chunk
#pragma unroll 4
        for (int k = 0; k < KC; k += 4) {
            v2f A, B;
            A.x = pA[ln * PA_S + k + 2 * half];
            A.y = pA[ln * PA_S + k + 2 * half + 1];
            B.x = hB[(k + 2 * half) * HB_S + 16 * wv + ln];
            B.y = hB[(k + 2 * half + 1) * HB_S + 16 * wv + ln];
            acc = __builtin_amdgcn_wmma_f32_16x16x4_f32(
                false, A, false, B, (short)0, acc, false, false);
        }
        __syncthreads();
    }

    // store 16x16 C tile: VGPR r -> (M = r + 8*half, N = 16*wv + ln)
#pragma unroll
    for (int r = 0; r < 8; ++r) {
        int mRow = r + 8 * half;
        out[(size_t)(i0 + mRow) * DIM + 16 * wv + ln] = acc[r];
    }
}

// ---------------------------------------------------------------------------
extern "C" void kernel_launch(void* const* d_in, const int* in_sizes, int n_in,
                              void* d_out, int out_size, void* d_ws, size_t ws_size,
                              hipStream_t stream) {
    const float* node  = (const float*)d_in[0];
    const float* adjacency = (const float*)d_in[1];
    const float* edge  = (const float*)d_in[2];
    const float* Wsrc  = (const float*)d_in[3];
    const float* bsrc  = (const float*)d_in[4];
    const float* Wdst  = (const float*)d_in[5];
    const float* bdst  = (const float*)d_in[6];
    const float* Wedge = (const float*)d_in[7];
    const float* bedge = (const float*)d_in[8];
    const float* a     = (const float*)d_in[9];

    float* out  = (float*)d_out;                    // 4096 x 128
    float* attn = out + (size_t)NN * DIM;           // 4096 x 4096

    float* ws = (float*)d_ws;
    float* h_dst = ws;                              // 4096*128
    float* s_src = h_dst + (size_t)NN * DIM;        // 4096
    float* s_dst = s_src + NN;                      // 4096
    float* w_ea  = s_dst + NN;                      // 9

    gat_prep<<<NN, DIM, 0, stream>>>(node, Wsrc, bsrc, Wdst, bdst,
                                     Wedge, bedge, a,
                                     h_dst, s_src, s_dst, w_ea);
    gat_attn<<<NN / 16, 256, 0, stream>>>(adjacency, edge, h_dst,
                                          s_src, s_dst, w_ea, out, attn);
}